// MultiBoxLoss_32873679684302
// MI455X (gfx1250) — compile-verified
//
#include <hip/hip_runtime.h>
#include <hip/hip_bf16.h>
#include <stdint.h>

typedef __attribute__((ext_vector_type(2))) float v2f;
typedef __attribute__((ext_vector_type(8))) float v8f;

namespace {
constexpr int kB = 64;
constexpr int kP = 8732;
constexpr int kC = 81;
constexpr int kT = 16;
constexpr int kTileRows = 128;
constexpr int kTileElems = kTileRows * kC;  // 10368 floats = 41472 B LDS
}

// ---------------------------------------------------------------------------
// Kernel 1: per-batch matching (one block per batch).
// Produces conf_t[B,P], num_pos[b], loc smooth-L1 sum[b]; zeroes ce accumulator.
// ---------------------------------------------------------------------------
__global__ __launch_bounds__(256) void mb_match_kernel(
    const float* __restrict__ loc, const float* __restrict__ priors,
    const float* __restrict__ truths, const int* __restrict__ labels,
    int* __restrict__ conf_t, int* __restrict__ np_ws,
    float* __restrict__ locl_ws, float* __restrict__ cep_ws) {
  __shared__ float sTr[kT * 4];
  __shared__ int sLab[kT];
  __shared__ unsigned long long sKey[kT];
  __shared__ int sBP[kT];
  __shared__ int sNP;
  __shared__ float sLL;
  const int b = blockIdx.x;
  const int tid = threadIdx.x;
  if (tid < kT) {
    float4 tb = reinterpret_cast<const float4*>(truths)[b * kT + tid];
    sTr[tid * 4 + 0] = tb.x; sTr[tid * 4 + 1] = tb.y;
    sTr[tid * 4 + 2] = tb.z; sTr[tid * 4 + 3] = tb.w;
    sLab[tid] = labels[b * kT + tid];
    sKey[tid] = 0ull;
  }
  if (tid == 0) { sNP = 0; sLL = 0.0f; cep_ws[b] = 0.0f; }
  __syncthreads();

  // Pass 1: per-truth argmax over priors (first occurrence via ~p tiebreak).
  unsigned long long key[kT];
#pragma unroll
  for (int t = 0; t < kT; ++t) key[t] = 0ull;
  for (int p = tid; p < kP; p += 256) {
    float4 pr = reinterpret_cast<const float4*>(priors)[p];
    float px0 = pr.x - 0.5f * pr.z, py0 = pr.y - 0.5f * pr.w;
    float px1 = pr.x + 0.5f * pr.z, py1 = pr.y + 0.5f * pr.w;
#pragma unroll
    for (int t = 0; t < kT; ++t) {
      float ix = fminf(sTr[t * 4 + 2], px1) - fmaxf(sTr[t * 4 + 0], px0);
      float iy = fminf(sTr[t * 4 + 3], py1) - fmaxf(sTr[t * 4 + 1], py0);
      ix = fmaxf(ix, 0.0f); iy = fmaxf(iy, 0.0f);
      float ov = ix * iy;  // overlaps >= 0 -> float bits are order-monotone
      unsigned long long k = ((unsigned long long)__float_as_uint(ov) << 32)
                           | (unsigned long long)(0xFFFFFFFFu - (unsigned)p);
      if (k > key[t]) key[t] = k;
    }
  }
#pragma unroll
  for (int t = 0; t < kT; ++t) atomicMax(&sKey[t], key[t]);
  __syncthreads();
  if (tid < kT) sBP[tid] = (int)(0xFFFFFFFFu - (unsigned)(sKey[tid] & 0xFFFFFFFFull));
  __syncthreads();

  // Pass 2: per-prior best truth + forcing + conf_t + loc loss.
  int npLocal = 0;
  float locSum = 0.0f;
  for (int p = tid; p < kP; p += 256) {
    float4 pr = reinterpret_cast<const float4*>(priors)[p];
    float px0 = pr.x - 0.5f * pr.z, py0 = pr.y - 0.5f * pr.w;
    float px1 = pr.x + 0.5f * pr.z, py1 = pr.y + 0.5f * pr.w;
    float bestOv = -1.0f; int bestT = 0;
#pragma unroll
    for (int t = 0; t < kT; ++t) {
      float ix = fminf(sTr[t * 4 + 2], px1) - fmaxf(sTr[t * 4 + 0], px0);
      float iy = fminf(sTr[t * 4 + 3], py1) - fmaxf(sTr[t * 4 + 1], py0);
      ix = fmaxf(ix, 0.0f); iy = fmaxf(iy, 0.0f);
      float ov = ix * iy;
      if (ov > bestOv) { bestOv = ov; bestT = t; }  // first-occurrence argmax
    }
#pragma unroll
    for (int j = 0; j < kT; ++j) {       // serial last-j-wins forcing
      if (sBP[j] == p) { bestT = j; bestOv = 2.0f; }
    }
    int cls = (bestOv < 0.5f) ? 0 : (sLab[bestT] + 1);
    conf_t[(size_t)b * kP + p] = cls;
    if (cls > 0) {
      ++npLocal;
      const float* m_ = &sTr[bestT * 4];
      float g0 = ((m_[0] + m_[2]) * 0.5f - pr.x) / (0.1f * pr.z);
      float g1 = ((m_[1] + m_[3]) * 0.5f - pr.y) / (0.1f * pr.w);
      float g2 = logf((m_[2] - m_[0]) / pr.z + 1e-10f) * 5.0f;  // /VAR[1]
      float g3 = logf((m_[3] - m_[1]) / pr.w + 1e-10f) * 5.0f;
      const float* ld = loc + ((size_t)b * kP + p) * 4;
      float g[4] = {g0, g1, g2, g3};
#pragma unroll
      for (int k2 = 0; k2 < 4; ++k2) {
        float d = ld[k2] - g[k2];
        float ad = fabsf(d);
        locSum += (ad < 1.0f) ? 0.5f * d * d : (ad - 0.5f);
      }
    }
  }
  atomicAdd(&sNP, npLocal);
  atomicAdd(&sLL, locSum);
  __syncthreads();
  if (tid == 0) { np_ws[b] = sNP; locl_ws[b] = sLL; }
}

// ---------------------------------------------------------------------------
// Kernel 2: fused log-softmax + CE, single streaming pass over conf_data.
// B128 async-to-LDS staging (tile base is 16B-aligned: (b*8732+128*bx)*324
// is a multiple of 16); WMMA (ones x exp-tile) does 16 row-sums per wave.
// Block = 256 threads = 8 waves; block tile = 128 rows x 81 classes.
// ---------------------------------------------------------------------------
__global__ __launch_bounds__(256) void mb_ce_kernel(
    const float* __restrict__ conf, const int* __restrict__ conf_t,
    float* __restrict__ mined, float* __restrict__ cep_ws) {
  __shared__ float sT[kTileElems];
  const int b = blockIdx.y;
  const int row0 = blockIdx.x * kTileRows;
  const int tid = threadIdx.x;
  int nrows = kP - row0; if (nrows > kTileRows) nrows = kTileRows;
  const int elemsValid = nrows * kC;

  // zero-fill pad rows (keeps WMMA inputs finite with EXEC all-ones)
  for (int i = elemsValid + tid; i < kTileElems; i += 256) sT[i] = 0.0f;

  // async copy global -> LDS (no VGPR round-trip), tracked by ASYNCcnt.
  // Main body in 16B chunks (512 B per wave-instruction), b32 tail.
  const float* gbase = conf + ((size_t)b * kP + row0) * kC;
  const unsigned ldsBase = (unsigned)(uintptr_t)(&sT[0]);
  const unsigned long long gBase64 = (unsigned long long)(uintptr_t)gbase;
  const int elemsV4 = elemsValid >> 2;  // float4 chunks
  for (int i = tid; i < elemsV4; i += 256) {
    unsigned ldsOff = ldsBase + (unsigned)(i * 16);
    unsigned long long ga = gBase64 + (unsigned long long)(i * 16);
    asm volatile("global_load_async_to_lds_b128 %0, %1, off"
                 :: "v"(ldsOff), "v"(ga) : "memory");
  }
  for (int i = (elemsV4 << 2) + tid; i < elemsValid; i += 256) {
    unsigned ldsOff = ldsBase + (unsigned)(i * 4);
    unsigned long long ga = gBase64 + (unsigned long long)(i * 4);
    asm volatile("global_load_async_to_lds_b32 %0, %1, off"
                 :: "v"(ldsOff), "v"(ga) : "memory");
  }
  asm volatile("s_wait_asynccnt 0x0" ::: "memory");
  __syncthreads();

  const int wave = tid >> 5;
  const int lane = tid & 31;
  const int n = lane & 15;        // row within this wave's 16-row slab
  const int half = lane >> 4;     // lane pair covers one row
  const float* rp = &sT[(wave * 16 + n) * kC];

  // row max: lane half scans odd/even classes, then one xor-16 shuffle
  float m = -3.0e38f;
  for (int k = half; k < kC; k += 2) m = fmaxf(m, rp[k]);
  m = fmaxf(m, __shfl_xor(m, 16, 32));

  // sum of exp via matrix pipe: D[m,n] = sum_k B[k,n] with A == ones(16x4).
  // Column n == row n of the slab; K-chunks of 4 classes, 21 chunks (pad->0).
  v2f a; a.x = 1.0f; a.y = 1.0f;
  v8f acc = {};
#pragma unroll
  for (int c4 = 0; c4 < 21; ++c4) {
    int kA = c4 * 4 + half * 2;
    int kB2 = kA + 1;
    float xa = rp[(kA  < kC) ? kA  : 0];
    float xb = rp[(kB2 < kC) ? kB2 : 0];
    v2f bv;
    bv.x = (kA  < kC) ? __expf(xa - m) : 0.0f;
    bv.y = (kB2 < kC) ? __expf(xb - m) : 0.0f;
    acc = __builtin_amdgcn_wmma_f32_16x16x4_f32(false, a, false, bv,
                                                (short)0, acc, false, false);
  }
  float sumexp = acc[0];  // every lane: row (lane&15) sum, replicated over M

  const int grow = row0 + wave * 16 + n;
  if (half == 0 && grow < kP) {
    int t = conf_t[(size_t)b * kP + grow];
    float ce = m + __logf(sumexp) - rp[t];
    ce = fmaxf(ce, 0.0f);  // ce >= 0 mathematically; kill fp noise for bit-order
    if (t > 0) atomicAdd(&cep_ws[b], ce);
    mined[(size_t)b * kP + grow] = (t > 0) ? 0.0f : ce;
  }
}

// ---------------------------------------------------------------------------
// Kernel 3: per-batch exact top-K sum of mined (K = min(3*num_pos, P-1)).
// Iterative strict-max scan; all mined values are >= 0 so bit order == order.
// ---------------------------------------------------------------------------
__global__ __launch_bounds__(256) void mb_topk_kernel(
    const float* __restrict__ mined, const int* __restrict__ np_ws,
    const float* __restrict__ cep_ws, float* __restrict__ lossc_ws) {
  __shared__ unsigned sMaxBits;
  __shared__ int sCnt;
  const int b = blockIdx.x;
  const int tid = threadIdx.x;
  const float* mv = mined + (size_t)b * kP;
  int K = np_ws[b] * 3;
  if (K > kP - 1) K = kP - 1;
  float bound = 3.0e38f;
  float sum = 0.0f;
  int taken = 0;
  while (taken < K) {
    if (tid == 0) { sMaxBits = 0u; sCnt = 0; }
    __syncthreads();
    unsigned lb = 0u;
    for (int p = tid; p < kP; p += 256) {
      float v = mv[p];
      if (v < bound) {
        unsigned bits = __float_as_uint(v);
        if (bits > lb) lb = bits;
      }
    }
    atomicMax(&sMaxBits, lb);
    __syncthreads();
    float m = __uint_as_float(sMaxBits);
    if (m <= 0.0f) break;  // remaining candidates are zeros
    int lc = 0;
    for (int p = tid; p < kP; p += 256) if (mv[p] == m) ++lc;
    atomicAdd(&sCnt, lc);
    __syncthreads();
    int cnt = sCnt;
    if (cnt <= 0) break;
    int take = (cnt < K - taken) ? cnt : (K - taken);
    sum += m * (float)take;
    taken += take;
    bound = m;
    __syncthreads();
  }
  if (tid == 0) lossc_ws[b] = cep_ws[b] + sum;
}

// ---------------------------------------------------------------------------
// Kernel 4: final 64-way reduction and normalization.
// ---------------------------------------------------------------------------
__global__ void mb_final_kernel(const int* __restrict__ np_ws,
                                const float* __restrict__ locl_ws,
                                const float* __restrict__ lossc_ws,
                                float* __restrict__ out) {
  if (threadIdx.x == 0 && blockIdx.x == 0) {
    int np = 0; float ll = 0.0f, lc = 0.0f;
    for (int b2 = 0; b2 < kB; ++b2) {
      np += np_ws[b2]; ll += locl_ws[b2]; lc += lossc_ws[b2];
    }
    float N = (float)((np < 1) ? 1 : np);
    out[0] = ll / N;
    out[1] = lc / N;
  }
}

extern "C" void kernel_launch(void* const* d_in, const int* in_sizes, int n_in,
                              void* d_out, int out_size, void* d_ws, size_t ws_size,
                              hipStream_t stream) {
  const float* loc    = (const float*)d_in[0];  // [B,P,4]
  const float* conf   = (const float*)d_in[1];  // [B,P,81]
  const float* priors = (const float*)d_in[2];  // [P,4]
  const float* truths = (const float*)d_in[3];  // [B,T,4]
  const int*   labels = (const int*)  d_in[4];  // [B,T]
  float* out = (float*)d_out;                   // [loss_l, loss_c]

  // workspace layout (~4.5 MB): conf_t[B*P] i32 | mined[B*P] f32 | scalars[B]x4
  char* ws = (char*)d_ws;
  const size_t BP = (size_t)kB * kP;
  int*   conf_t   = (int*)  (ws);
  float* mined    = (float*)(ws + BP * 4);
  int*   np_ws    = (int*)  (ws + BP * 8);
  float* locl_ws  = (float*)(ws + BP * 8 + kB * 4);
  float* cep_ws   = (float*)(ws + BP * 8 + kB * 8);
  float* lossc_ws = (float*)(ws + BP * 8 + kB * 12);

  mb_match_kernel<<<kB, 256, 0, stream>>>(loc, priors, truths, labels,
                                          conf_t, np_ws, locl_ws, cep_ws);
  dim3 g2((kP + kTileRows - 1) / kTileRows, kB);
  mb_ce_kernel<<<g2, 256, 0, stream>>>(conf, conf_t, mined, cep_ws);
  mb_topk_kernel<<<kB, 256, 0, stream>>>(mined, np_ws, cep_ws, lossc_ws);
  mb_final_kernel<<<1, 1, 0, stream>>>(np_ws, locl_ws, lossc_ws, out);
}